// BodyAgnosticNACPG_19387482375000
// MI455X (gfx1250) — compile-verified
//
#include <hip/hip_runtime.h>
#include <math.h>

// Rank-2 decomposition of the Kuramoto-style coupling:
//   sum_x[i] = cos(p_i)*A + sin(p_i)*B
//   sum_y[i] = sin(p_i)*A - cos(p_i)*B
// with A = sum_j cos(p_j)x_j + sin(p_j)y_j, B = sum_j sin(p_j)x_j - cos(p_j)y_j.
// O(N) instead of O(N^2); memory-bound (~360KB total traffic @ 23.3 TB/s).

typedef __attribute__((ext_vector_type(2))) float v2f;
typedef __attribute__((ext_vector_type(8))) float v8f;

#define COUP  0.08f
#define ALPHA 0.45f
#define DT    0.01f
#define DIFF  10.0f

// Exact fp32 sum of one value per lane across a wave32 using
// V_WMMA_F32_16X16X4_F32 with an all-ones B matrix.
// A layout (16x4 f32): lane L<16 holds (M=L, K=0/1) in its 2 VGPRs,
// lane L>=16 holds (M=L-16, K=2/3). We put the partial in VGPR0, 0 in VGPR1,
// so rowsum(m) = a_m + a_{m+16}. With B = ones, every D column equals the
// rowsums; summing a lane's 8 D VGPRs gives half the rows, and one xor-16
// shuffle-add completes the 32-lane total (identical in all lanes).
__device__ __forceinline__ float wave_reduce_wmma(float v) {
    v2f A;  A.x = v;    A.y = 0.0f;
    v2f B;  B.x = 1.0f; B.y = 1.0f;   // all-ones B: layout-invariant
    v8f C = {};
    v8f D = __builtin_amdgcn_wmma_f32_16x16x4_f32(
        /*neg_a=*/false, A, /*neg_b=*/false, B,
        /*c_mod=*/(short)0, C, /*reuse_a=*/false, /*reuse_b=*/false);
    float half_sum = D[0] + D[1] + D[2] + D[3] + D[4] + D[5] + D[6] + D[7];
    half_sum += __shfl_xor(half_sum, 16, 32);
    return half_sum;   // full 32-lane sum, broadcast to all lanes
}

__global__ __launch_bounds__(1024)
void nacpg_reduce_AB(const float* __restrict__ phase,
                     const float* __restrict__ xy,
                     float* __restrict__ ws, int n) {
    __shared__ float sA[32];
    __shared__ float sB[32];

    const int tid = threadIdx.x;
    float a = 0.0f, b = 0.0f;
    for (int j = tid; j < n; j += 1024) {
        float p = phase[j];
        float c, s;
        __sincosf(p, &s, &c);
        // use accurate versions to match the fp32 reference closely
        s = sinf(p);
        c = cosf(p);
        float x = xy[2 * j];
        float y = xy[2 * j + 1];
        a = fmaf(c, x, fmaf(s, y, a));
        b = fmaf(s, x, fmaf(-c, y, b));
    }

    // per-wave reduction via WMMA (EXEC is all-ones: uniform control flow)
    float wa = wave_reduce_wmma(a);
    float wb = wave_reduce_wmma(b);

    const int wave = tid >> 5;
    const int lane = tid & 31;
    if (lane == 0) { sA[wave] = wa; sB[wave] = wb; }
    __syncthreads();

    if (tid == 0) {
        float A = 0.0f, Bt = 0.0f;
        const int nwaves = blockDim.x >> 5;
        for (int i = 0; i < nwaves; ++i) { A += sA[i]; Bt += sB[i]; }
        ws[0] = A;
        ws[1] = Bt;
    }
}

__global__ void nacpg_update(const float* __restrict__ phase,
                             const float* __restrict__ w,
                             const float* __restrict__ xy,
                             const float* __restrict__ xyd_old,
                             const float* __restrict__ ws,
                             float* __restrict__ out, int n) {
    const int i = blockIdx.x * blockDim.x + threadIdx.x;
    if (i >= n) return;

    const float A = ws[0];
    const float B = ws[1];

    const float p  = phase[i];
    const float ci = cosf(p);
    const float si = sinf(p);

    const float x = xy[2 * i];
    const float y = xy[2 * i + 1];

    const float sum_x = fmaf(ci, A,  si * B);
    const float sum_y = fmaf(si, A, -ci * B);

    const float inv_deg = 1.0f / (float)(n - 1);
    const float coup_x = COUP * (sum_x - x) * inv_deg;
    const float coup_y = COUP * (sum_y - y) * inv_deg;

    const float r2 = fmaf(x, x, y * y);
    const float a  = ALPHA * (1.0f - r2);
    const float wi = w[i];

    float xdot = fmaf(a, x, fmaf(-wi, y, coup_x));
    float ydot = fmaf(wi, x, fmaf(a, y, coup_y));

    const float xo = xyd_old[2 * i];
    const float yo = xyd_old[2 * i + 1];
    xdot = fminf(fmaxf(xdot, xo - DIFF), xo + DIFF);
    ydot = fminf(fmaxf(ydot, yo - DIFF), yo + DIFF);

    const float nx = fmaf(DT, xdot, x);
    const float ny = fmaf(DT, ydot, y);

    // out layout: [0,2N) new_xy | [2N,4N) new_xyd | [4N,5N) y_unit
    out[2 * i]             = nx;
    out[2 * i + 1]         = ny;
    out[2 * n + 2 * i]     = xdot;
    out[2 * n + 2 * i + 1] = ydot;
    out[4 * n + i]         = fminf(fmaxf(ny, -1.0f), 1.0f);
}

extern "C" void kernel_launch(void* const* d_in, const int* in_sizes, int n_in,
                              void* d_out, int out_size, void* d_ws, size_t ws_size,
                              hipStream_t stream) {
    const float* phase   = (const float*)d_in[0];
    const float* w       = (const float*)d_in[1];
    const float* xy      = (const float*)d_in[2];
    const float* xyd_old = (const float*)d_in[3];
    float* out = (float*)d_out;
    float* ws  = (float*)d_ws;
    const int n = in_sizes[0];

    nacpg_reduce_AB<<<1, 1024, 0, stream>>>(phase, xy, ws, n);
    nacpg_update<<<(n + 255) / 256, 256, 0, stream>>>(phase, w, xy, xyd_old, ws, out, n);
}